// Attention_21002390077921
// MI455X (gfx1250) — compile-verified
//
#include <hip/hip_runtime.h>

// ---------------------------------------------------------------------------
// Paged-attention decode (GQA) for MI455X / gfx1250, fp32 end-to-end.
// WMMA: V_WMMA_F32_16X16X4_F32 chained over head_dim (QK^T) and tokens (PV).
// Next-tile K/V rows prefetched (global_prefetch_b8) to overlap HBM latency
// with the WMMA/softmax phase.
// ---------------------------------------------------------------------------

#define SCALE 0.08838834764831845f
#define NUM_HEADS 32
#define KVH 8
#define HD 128
#define BS 16
#define NUM_BLOCKS 4096
#define MBPS 128
#define BATCH 32
#define DTOT (KVH * HD)          // 1024
#define NSLOTS (NUM_BLOCKS * BS) // 65536
#define SPLITS 16
#define TILES_PER_SPLIT (MBPS / SPLITS) // 8
#define PART_STRIDE 520                 // 4 m + 4 l + 4*128 O

#define KSTR 132 // LDS row stride for K/V tiles (bank-conflict-free b64 reads)
#define PSTR 18  // LDS row stride for P (even -> 8B aligned b64, conflict-free)

typedef float v2f __attribute__((ext_vector_type(2)));
typedef float v4f __attribute__((ext_vector_type(4)));
typedef float v8f __attribute__((ext_vector_type(8)));

// ---------------------------------------------------------------------------
// Kernel 0: build override table (flat slot -> batch index or -1) so we never
// mutate the input k_cache/v_cache (reference scatters; we redirect on gather).
// ---------------------------------------------------------------------------
__global__ __launch_bounds__(256) void build_override_kernel(
    const int* __restrict__ slot_mapping, int* __restrict__ ovr) {
  const int t = blockIdx.x * blockDim.x + threadIdx.x;
  int ov = -1;
#pragma unroll
  for (int j = 0; j < BATCH; ++j)
    if (slot_mapping[j] == t) ov = j;
  ovr[t] = ov;
}

// ---------------------------------------------------------------------------
// Kernel 1: one wave32 per (split, kvh, batch). Online-softmax over its tile
// range, writes unnormalized partial (m[4], l[4], O[4][128]) to workspace.
// ---------------------------------------------------------------------------
__global__ __launch_bounds__(32) void paged_attn_kernel(
    const float* __restrict__ q, const float* __restrict__ knew,
    const float* __restrict__ vnew, const float* __restrict__ kcache,
    const float* __restrict__ vcache, const int* __restrict__ ovr,
    const int* __restrict__ btab, const int* __restrict__ ctx_lens,
    float* __restrict__ partials) {
  const int split = blockIdx.x;
  const int kvh = blockIdx.y;
  const int b = blockIdx.z;
  const int lane = threadIdx.x;
  const int hl = lane & 15;          // row/col index within 16-lane half
  const int off = (lane >> 4) << 1;  // 0 for lanes 0-15, 2 for lanes 16-31

  __shared__ float sK[16 * KSTR];
  __shared__ float sV[16 * KSTR];
  __shared__ float sP[16 * PSTR];

  const int ctx = ctx_lens[b];
  const int nT = (ctx + BS - 1) >> 4;
  const int t0 = split * TILES_PER_SPLIT;
  const int t1 = min(t0 + TILES_PER_SPLIT, nT);

  // --- Q fragments, pre-scaled by 1/sqrt(128). A-layout for 16x16x4 f32:
  // lane<16:  x=Q[m=lane][4c+0], y=Q[m][4c+1]
  // lane>=16: x=Q[m=lane-16][4c+2], y=Q[m][4c+3]; rows 4..15 zero padding.
  v2f aq[32];
  {
    const float* qrow =
        q + (((size_t)b * NUM_HEADS + (kvh << 2) + hl) << 7);
#pragma unroll
    for (int c = 0; c < 32; ++c) {
      v2f tq;
      tq.x = 0.f;
      tq.y = 0.f;
      if (hl < 4) tq = *(const v2f*)(qrow + 4 * c + off);
      aq[c].x = tq.x * SCALE;
      aq[c].y = tq.y * SCALE;
    }
  }

  v8f acc_o[8];
  float m_r[8], l_r[8];
#pragma unroll
  for (int dc = 0; dc < 8; ++dc)
#pragma unroll
    for (int e = 0; e < 8; ++e) acc_o[dc][e] = 0.f;
#pragma unroll
  for (int j = 0; j < 8; ++j) {
    m_r[j] = -1e30f;
    l_r[j] = 0.f;
  }

  for (int t = t0; t < t1; ++t) {
    const int blk = btab[b * MBPS + t];

    // ---- stage K/V tile: one perfectly-coalesced 512B row per instruction
#pragma unroll 4
    for (int r = 0; r < 16; ++r) {
      const int flat = blk * BS + r;
      const int ov = ovr[flat];
      const float* kr;
      const float* vr;
      if (ov >= 0) {  // slot overwritten by this decode step's new K/V token
        kr = knew + ((size_t)ov * KVH + kvh) * HD;
        vr = vnew + ((size_t)ov * KVH + kvh) * HD;
      } else {
        kr = kcache + (size_t)flat * DTOT + kvh * HD;
        vr = vcache + (size_t)flat * DTOT + kvh * HD;
      }
      const v4f kq = *(const v4f*)(kr + 4 * lane);
      const v4f vq = *(const v4f*)(vr + 4 * lane);
      *(v4f*)(sK + r * KSTR + 4 * lane) = kq;
      *(v4f*)(sV + r * KSTR + 4 * lane) = vq;
    }

    // ---- prefetch next tile's K/V rows into cache (global_prefetch_b8);
    // overlaps the whole WMMA/softmax phase with the next HBM fetch. No
    // override check needed: prefetching a stale line is harmless.
    if (t + 1 < t1) {
      const int blkn = btab[b * MBPS + t + 1];
#pragma unroll 4
      for (int r = 0; r < 16; ++r) {
        const size_t base = (size_t)(blkn * BS + r) * DTOT + kvh * HD;
        __builtin_prefetch(kcache + base + 4 * lane, 0, 3);
        __builtin_prefetch(vcache + base + 4 * lane, 0, 3);
      }
    }

    // ---- S[16q,16t] = Q * K^T : 32 chained f32 WMMAs over head_dim
    // B-layout: lane<16: x=K[tok=lane][4c+0], y=[4c+1]; lane>=16: +2/+3.
    v8f s;
#pragma unroll
    for (int e = 0; e < 8; ++e) s[e] = 0.f;
#pragma unroll
    for (int c = 0; c < 32; ++c) {
      const v2f bk = *(const v2f*)(sK + hl * KSTR + 4 * c + off);
      s = __builtin_amdgcn_wmma_f32_16x16x4_f32(false, aq[c], false, bk,
                                                (short)0, s, false, false);
    }

    // ---- online softmax (rows live in C/D layout: elem j <-> row j / j+8)
    const bool valid = (t * BS + hl) < ctx;
    float pr[8], alpha[8];
#pragma unroll
    for (int j = 0; j < 8; ++j) {
      const float sv = valid ? s[j] : -1e30f;
      float mx = sv;
      mx = fmaxf(mx, __shfl_xor(mx, 1, 32));
      mx = fmaxf(mx, __shfl_xor(mx, 2, 32));
      mx = fmaxf(mx, __shfl_xor(mx, 4, 32));
      mx = fmaxf(mx, __shfl_xor(mx, 8, 32));
      const float mn = fmaxf(m_r[j], mx);
      const float a = __expf(m_r[j] - mn);
      const float p = __expf(sv - mn);
      float rs = p;
      rs += __shfl_xor(rs, 1, 32);
      rs += __shfl_xor(rs, 2, 32);
      rs += __shfl_xor(rs, 4, 32);
      rs += __shfl_xor(rs, 8, 32);
      l_r[j] = l_r[j] * a + rs;
      m_r[j] = mn;
      alpha[j] = a;
      pr[j] = p;
    }
#pragma unroll
    for (int dc = 0; dc < 8; ++dc)
#pragma unroll
      for (int j = 0; j < 8; ++j) acc_o[dc][j] *= alpha[j];

    // ---- P (C/D layout) -> LDS row-major, padded stride
    const int prow_base = (lane >> 4) << 3;
#pragma unroll
    for (int j = 0; j < 8; ++j) sP[(prow_base + j) * PSTR + hl] = pr[j];

    // ---- O[16,128] += P[16,16] * V[16,128] : 8 d-chunks x 4 k-chunks
    v2f aP[4];
#pragma unroll
    for (int kc = 0; kc < 4; ++kc)
      aP[kc] = *(const v2f*)(sP + hl * PSTR + 4 * kc + off);
#pragma unroll
    for (int dc = 0; dc < 8; ++dc) {
#pragma unroll
      for (int kc = 0; kc < 4; ++kc) {
        const int vrow = 4 * kc + off;
        v2f bv;
        bv.x = sV[vrow * KSTR + dc * BS + hl];
        bv.y = sV[(vrow + 1) * KSTR + dc * BS + hl];
        acc_o[dc] = __builtin_amdgcn_wmma_f32_16x16x4_f32(
            false, aP[kc], false, bv, (short)0, acc_o[dc], false, false);
      }
    }
  }

  // ---- write unnormalized partial (rows 0..3 only are real q-heads)
  float* part =
      partials + (((size_t)b * KVH + kvh) * SPLITS + split) * PART_STRIDE;
  if (lane == 0) {
#pragma unroll
    for (int j = 0; j < 4; ++j) {
      part[j] = m_r[j];
      part[4 + j] = l_r[j];
    }
  }
  if (lane < 16) {
#pragma unroll
    for (int dc = 0; dc < 8; ++dc)
#pragma unroll
      for (int j = 0; j < 4; ++j)
        part[8 + j * HD + dc * BS + hl] = acc_o[dc][j];
  }
}

// ---------------------------------------------------------------------------
// Kernel 2: log-sum-exp combine of the 16 context splits, normalize, write out
// ---------------------------------------------------------------------------
__global__ __launch_bounds__(128) void combine_kernel(
    const float* __restrict__ partials, float* __restrict__ out) {
  const int kvh = blockIdx.x;
  const int b = blockIdx.y;
  const float* base =
      partials + ((size_t)b * KVH + kvh) * SPLITS * PART_STRIDE;
  for (int i = threadIdx.x; i < 4 * HD; i += blockDim.x) {
    const int row = i >> 7;
    const int d = i & (HD - 1);
    float M = -1e30f;
#pragma unroll
    for (int s = 0; s < SPLITS; ++s)
      M = fmaxf(M, base[s * PART_STRIDE + row]);
    float L = 0.f, O = 0.f;
#pragma unroll
    for (int s = 0; s < SPLITS; ++s) {
      const float e = __expf(base[s * PART_STRIDE + row] - M);
      L += base[s * PART_STRIDE + 4 + row] * e;
      O += base[s * PART_STRIDE + 8 + i] * e;
    }
    out[((size_t)b * NUM_HEADS + (kvh << 2) + row) * HD + d] = O / L;
  }
}

// ---------------------------------------------------------------------------
extern "C" void kernel_launch(void* const* d_in, const int* in_sizes, int n_in,
                              void* d_out, int out_size, void* d_ws,
                              size_t ws_size, hipStream_t stream) {
  (void)in_sizes; (void)n_in; (void)out_size; (void)ws_size;
  const float* q = (const float*)d_in[0];
  const float* k = (const float*)d_in[1];
  const float* v = (const float*)d_in[2];
  const float* kcache = (const float*)d_in[3];
  const float* vcache = (const float*)d_in[4];
  const int* slot = (const int*)d_in[5];
  const int* btab = (const int*)d_in[6];
  const int* clen = (const int*)d_in[7];
  float* out = (float*)d_out;

  // workspace layout: [65536 x i32 override] [32*8*16 x 520 f32 partials]
  int* ovr = (int*)d_ws;
  float* partials = (float*)((char*)d_ws + NSLOTS * sizeof(int));

  build_override_kernel<<<NSLOTS / 256, 256, 0, stream>>>(slot, ovr);
  dim3 g1(SPLITS, KVH, BATCH);
  paged_attn_kernel<<<g1, 32, 0, stream>>>(q, k, v, kcache, vcache, ovr, btab,
                                           clen, partials);
  dim3 g2(KVH, BATCH);
  combine_kernel<<<g2, 128, 0, stream>>>(partials, out);
}